// MultiHeadGATLayerWithEdgeFeatures_57200374448375
// MI455X (gfx1250) — compile-verified
//
#include <hip/hip_runtime.h>
#include <hip/hip_bf16.h>

typedef __attribute__((ext_vector_type(2))) float v2f;
typedef __attribute__((ext_vector_type(8))) float v8f;

#define N_NODES 4096
#define IN_F    256
#define OUT_F   64
#define HEADS   4
#define EDGE_F  32
#define ALPHA   0.2f

// ---------------------------------------------------------------------------
// CDNA5 async global->LDS copy (ASYNCcnt path), per CDNA5 ISA
// VDST = 32-bit LDS byte address, VADDR = 64-bit global address.
// ---------------------------------------------------------------------------
__device__ __forceinline__ void async_copy_b128(void* lds_dst, const void* gsrc) {
    unsigned lds = (unsigned)(unsigned long long)
        (__attribute__((address_space(3))) void*)lds_dst;
    asm volatile("global_load_async_to_lds_b128 %0, %1, off"
                 :: "v"(lds), "v"(gsrc)
                 : "memory");
}
__device__ __forceinline__ void wait_asynccnt0() {
    asm volatile("s_wait_asynccnt 0" ::: "memory");
}

// ---------------------------------------------------------------------------
// Kernel 1: Wh = h @ W   (4096x256 @ 256x256, fp32 WMMA 16x16x4)
// One wave per 16x16 output tile. grid = (256 i-tiles, 16 n-tiles), block=32.
// ---------------------------------------------------------------------------
__global__ void wh_gemm_kernel(const float* __restrict__ h,
                               const float* __restrict__ W,
                               float* __restrict__ Wh) {
    const int lane = threadIdx.x;
    const int it   = blockIdx.x;
    const int nt   = blockIdx.y;
    const int m    = lane & 15;
    const int kh   = (lane >> 4) << 1;
    const int hf   = lane >> 4;

    const float* hrow = h + (size_t)(it * 16 + m) * IN_F;
    const float* wcol = W + nt * 16 + m;

    v8f c = {};
    for (int k = 0; k < IN_F; k += 4) {
        v2f a, b;
        a.x = hrow[k + kh + 0];
        a.y = hrow[k + kh + 1];
        b.x = wcol[(size_t)(k + kh + 0) * (HEADS * OUT_F)];
        b.y = wcol[(size_t)(k + kh + 1) * (HEADS * OUT_F)];
        c = __builtin_amdgcn_wmma_f32_16x16x4_f32(false, a, false, b,
                                                  (short)0, c, false, false);
    }
#pragma unroll
    for (int v = 0; v < 8; ++v) {
        int row = it * 16 + v + hf * 8;
        Wh[(size_t)row * (HEADS * OUT_F) + nt * 16 + m] = c[v];
    }
}

// ---------------------------------------------------------------------------
// Kernel 2: per-node attention coefficients
// ---------------------------------------------------------------------------
__global__ void coeff_kernel(const float* __restrict__ Wh,
                             const float* __restrict__ edge_attr,
                             const float* __restrict__ a,
                             float* __restrict__ srcb,
                             float* __restrict__ cdstb) {
    int gid = blockIdx.x * blockDim.x + threadIdx.x;
    if (gid >= N_NODES * HEADS) return;
    int n  = gid >> 2;
    int hh = gid & 3;
    const float* whp = Wh + (size_t)n * (HEADS * OUT_F) + hh * OUT_F;
    const float* av  = a + hh * (2 * OUT_F + EDGE_F);
    float s = 0.f, d = 0.f;
#pragma unroll 8
    for (int o = 0; o < OUT_F; ++o) {
        float w = whp[o];
        s += w * av[o];
        d += w * av[OUT_F + o];
    }
    const float* ea = edge_attr + (size_t)n * EDGE_F;
#pragma unroll 8
    for (int e = 0; e < EDGE_F; ++e) d += ea[e] * av[2 * OUT_F + e];
    srcb[gid]  = s;
    cdstb[gid] = d;
}

// ---------------------------------------------------------------------------
// Kernel 3: fused masked-softmax attention + aggregation.
// Single unnormalized pass; adj & cdst tiles double-buffered in LDS via
// async global->LDS b128 copies overlapped with fp32 WMMA compute.
// Block = 16 output rows, 4 waves = 4 heads.
// ---------------------------------------------------------------------------
__global__ void gat_agg_kernel(const int*   __restrict__ adj,
                               const float* __restrict__ Wh,
                               const float* __restrict__ srcb,
                               const float* __restrict__ cdstb,
                               float* __restrict__ out) {
    __shared__ int   s_adj[2][16 * 64];      // 2 x 4KB
    __shared__ float s_c[2][64 * HEADS];     // 2 x 1KB (contiguous copy of cdst chunk)
    __shared__ float s_den[HEADS][16];

    const int tid  = threadIdx.x;            // 0..127
    const int lane = tid & 31;
    const int head = tid >> 5;               // wave id == head
    const int ib   = blockIdx.x * 16;
    const int m    = lane & 15;
    const int kh   = (lane >> 4) << 1;       // 0 or 2
    const int hf   = lane >> 4;

    const float srci = srcb[(size_t)(ib + m) * HEADS + head];

    v8f acc[4] = {v8f{}, v8f{}, v8f{}, v8f{}};
    float dpart = 0.f;

    // async-stage one 64-column chunk (adj 16x64 tile + cdst 64x4) into buffer b
    auto stage = [&](int j0, int b) {
#pragma unroll
        for (int r = 0; r < 2; ++r) {
            int idx16 = r * 128 + tid;        // 256 x b128 units cover 16x64 ints
            int rr = idx16 >> 4;              // row 0..15
            int c4 = (idx16 & 15) << 2;       // col 0..60 step 4
            async_copy_b128(&s_adj[b][rr * 64 + c4],
                            &adj[(size_t)(ib + rr) * N_NODES + j0 + c4]);
        }
        if (tid < 64)                         // cdst chunk is 1KB contiguous
            async_copy_b128(&s_c[b][tid * 4],
                            &cdstb[(size_t)j0 * HEADS + tid * 4]);
    };

    stage(0, 0);

    int buf = 0;
    for (int j0 = 0; j0 < N_NODES; j0 += 64) {
        wait_asynccnt0();     // this wave's async copies for chunk j0 done
        __syncthreads();      // all waves' copies visible; prev compute finished
        if (j0 + 64 < N_NODES) stage(j0 + 64, buf ^ 1);   // overlap next chunk

        const int*   adj_t  = s_adj[buf];
        const float* c_t    = s_c[buf];
        const float* whbase = Wh + (size_t)j0 * (HEADS * OUT_F) + head * OUT_F;
        __builtin_prefetch(whbase + (size_t)64 * (HEADS * OUT_F), 0, 0);

        for (int kk = 0; kk < 64; kk += 4) {
            // A operand: attention weights built on the fly
            v2f aw;
#pragma unroll
            for (int v = 0; v < 2; ++v) {
                int jj  = kk + kh + v;
                float x = srci + c_t[jj * HEADS + head];
                float e = x > 0.f ? x : ALPHA * x;
                float w = (adj_t[m * 64 + jj] > 0) ? __expf(e) : 1.0f;
                dpart += w;
                aw[v] = w;
            }
            // B operand: Wh rows j0+kk+kh, +1; this head's 64 columns
            const float* w0 = whbase + (size_t)(kk + kh) * (HEADS * OUT_F);
            const float* w1 = w0 + (HEADS * OUT_F);
            v2f b0, b1, b2, b3;
            b0.x = w0[m];        b0.y = w1[m];
            b1.x = w0[16 + m];   b1.y = w1[16 + m];
            b2.x = w0[32 + m];   b2.y = w1[32 + m];
            b3.x = w0[48 + m];   b3.y = w1[48 + m];
            acc[0] = __builtin_amdgcn_wmma_f32_16x16x4_f32(false, aw, false, b0, (short)0, acc[0], false, false);
            acc[1] = __builtin_amdgcn_wmma_f32_16x16x4_f32(false, aw, false, b1, (short)0, acc[1], false, false);
            acc[2] = __builtin_amdgcn_wmma_f32_16x16x4_f32(false, aw, false, b2, (short)0, acc[2], false, false);
            acc[3] = __builtin_amdgcn_wmma_f32_16x16x4_f32(false, aw, false, b3, (short)0, acc[3], false, false);
        }
        buf ^= 1;
    }

    // denom: lane(m,half) holds half of row m's sum; fold halves, broadcast via LDS
    dpart += __shfl_xor(dpart, 16, 32);
    if (lane < 16) s_den[head][lane] = dpart;
    __syncthreads();

    // Normalize + ELU + store.  D layout: row = v + 8*hf, col = m
#pragma unroll
    for (int t = 0; t < 4; ++t) {
#pragma unroll
        for (int v = 0; v < 8; ++v) {
            int row   = v + hf * 8;
            float val = acc[t][v] / s_den[head][row];
            val = val > 0.f ? val : (__expf(val) - 1.f);
            out[(size_t)(ib + row) * (HEADS * OUT_F) + head * OUT_F + t * 16 + m] = val;
        }
    }
}

extern "C" void kernel_launch(void* const* d_in, const int* in_sizes, int n_in,
                              void* d_out, int out_size, void* d_ws, size_t ws_size,
                              hipStream_t stream) {
    const float* h         = (const float*)d_in[0];
    const int*   adj       = (const int*)  d_in[1];
    const float* edge_attr = (const float*)d_in[2];
    const float* W         = (const float*)d_in[3];
    const float* a         = (const float*)d_in[4];
    float* out = (float*)d_out;

    float* Wh    = (float*)d_ws;                        // 4 MB
    float* srcb  = Wh + (size_t)N_NODES * HEADS * OUT_F;
    float* cdstb = srcb + (size_t)N_NODES * HEADS;

    wh_gemm_kernel<<<dim3(N_NODES / 16, (HEADS * OUT_F) / 16), 32, 0, stream>>>(h, W, Wh);
    coeff_kernel<<<(N_NODES * HEADS + 255) / 256, 256, 0, stream>>>(Wh, edge_attr, a, srcb, cdstb);
    gat_agg_kernel<<<N_NODES / 16, 128, 0, stream>>>(adj, Wh, srcb, cdstb, out);
}